// BNAFModel_57930518888500
// MI455X (gfx1250) — compile-verified
//
#include <hip/hip_runtime.h>
#include <hip/hip_bf16.h>
#include <math.h>

// BNAF density evaluation for MI455X (gfx1250), fp32 WMMA path.
// GEMMs run on v_wmma_f32_16x16x4_f32 with LDS-staged, double-buffered tiles:
// chunk i+1 global_load_b128s are issued before chunk i's 32 WMMAs so the
// matrix pipe is not stalled on s_wait_loadcnt per instruction, and the B tile
// is fetched once per block (shared by 4 waves via LDS) instead of per wave.

typedef __attribute__((ext_vector_type(2))) float v2f;
typedef __attribute__((ext_vector_type(4))) float v4f;
typedef __attribute__((ext_vector_type(8))) float v8f;

#define DIMV 64
#define DH   2048      // DIM*HID
#define BV   1024
#define KC   32        // K-chunk staged in LDS
#define SRD  36        // LDS row stride (floats): 36 -> conflict-free ds_load_b64

__device__ __forceinline__ float softplusf(float z) {
    return fmaxf(z, 0.0f) + log1pf(expf(-fabsf(z)));
}
// log(1 - tanh(x)^2), computed stably (matches reference formula)
__device__ __forceinline__ float tanh_lgrad(float x) {
    return -2.0f * (x - 0.69314718055994530942f + softplusf(-2.0f * x));
}

// ---------------- Weight preparation ----------------
// Layer 0: W (2048 x 64), ob=32, ib=1. One 64-thread block per row.
__global__ void prep_w0(const float* __restrict__ W, const float* __restrict__ dw,
                        float* __restrict__ wn, float* __restrict__ g0) {
    __shared__ float red[64];
    int r = blockIdx.x;          // 0..2047
    int c = threadIdx.x;         // 0..63
    int i = r >> 5;              // diagonal block index
    float Wv = W[r * 64 + c];
    float w = (c == i) ? expf(Wv) : ((c < i) ? Wv : 0.0f);
    red[c] = w * w;
    __syncthreads();
    for (int s = 32; s > 0; s >>= 1) { if (c < s) red[c] += red[c + s]; __syncthreads(); }
    float wsn = red[0];
    float scale = expf(dw[r]) * rsqrtf(wsn);
    wn[r * 64 + c] = w * scale;
    if (c == 0) g0[r] = dw[r] + W[r * 64 + i] - 0.5f * logf(wsn);
}

// Layer 1: W (2048 x 2048), ob=ib=32. One 256-thread block per row.
__global__ void prep_w1(const float* __restrict__ W, const float* __restrict__ dw,
                        float* __restrict__ wn, float* __restrict__ g1) {
    __shared__ float red[256];
    int r = blockIdx.x;
    int t = threadIdx.x;
    int i = r >> 5;
    int diag0 = i << 5;
    int lim = diag0 + 32;
    const float* Wr = W + (size_t)r * DH;
    float acc = 0.0f;
    for (int c = t; c < DH; c += 256) {
        float Wv = Wr[c];
        float w = (c >= lim) ? 0.0f : ((c >= diag0) ? expf(Wv) : Wv);
        acc += w * w;
    }
    red[t] = acc;
    __syncthreads();
    for (int s = 128; s > 0; s >>= 1) { if (t < s) red[t] += red[t + s]; __syncthreads(); }
    float wsn = red[0];
    float scale = expf(dw[r]) * rsqrtf(wsn);
    float* wnr = wn + (size_t)r * DH;
    for (int c = t; c < DH; c += 256) {
        float Wv = Wr[c];
        float w = (c >= lim) ? 0.0f : ((c >= diag0) ? expf(Wv) : Wv);
        wnr[c] = w * scale;
    }
    if (t < 32) {
        int a = r & 31;
        g1[i * 1024 + a * 32 + t] = dw[r] + Wr[diag0 + t] - 0.5f * logf(wsn);
    }
}

// Layer 2: W (64 x 2048), ob=1, ib=32. One 256-thread block per row.
__global__ void prep_w2(const float* __restrict__ W, const float* __restrict__ dw,
                        float* __restrict__ wn, float* __restrict__ g2) {
    __shared__ float red[256];
    int r = blockIdx.x;          // 0..63 == block index i
    int t = threadIdx.x;
    int diag0 = r << 5;
    int lim = diag0 + 32;
    const float* Wr = W + (size_t)r * DH;
    float acc = 0.0f;
    for (int c = t; c < DH; c += 256) {
        float Wv = Wr[c];
        float w = (c >= lim) ? 0.0f : ((c >= diag0) ? expf(Wv) : Wv);
        acc += w * w;
    }
    red[t] = acc;
    __syncthreads();
    for (int s = 128; s > 0; s >>= 1) { if (t < s) red[t] += red[t + s]; __syncthreads(); }
    float wsn = red[0];
    float scale = expf(dw[r]) * rsqrtf(wsn);
    float* wnr = wn + (size_t)r * DH;
    for (int c = t; c < DH; c += 256) {
        float Wv = Wr[c];
        float w = (c >= lim) ? 0.0f : ((c >= diag0) ? expf(Wv) : Wv);
        wnr[c] = w * scale;
    }
    if (t < 32) g2[r * 32 + t] = dw[r] + Wr[diag0 + t] - 0.5f * logf(wsn);
}

// ---------------- WMMA GEMM: C(MxN) = A(MxK) * Wn(NxK)^T + bias ----------------
// block = 128 threads (4 waves). Block tile: M=64 (wave -> 16 rows) x N=64.
// K staged through LDS in KC=32 chunks, double-buffered.
// mode 0: out_act = tanh(pre), out_grad = g0[col] + log(1-tanh^2)
// mode 1: out_act = tanh(pre), out_grad = log(1-tanh^2)
// mode 2: out_act = pre
__global__ void wmma_gemm(const float* __restrict__ A, const float* __restrict__ Wn,
                          const float* __restrict__ bias, int K, int N, int mode,
                          float* __restrict__ out_act, float* __restrict__ out_grad,
                          const float* __restrict__ g0) {
    __shared__ float ldsA[2][64 * SRD];
    __shared__ float ldsB[2][64 * SRD];

    int tid  = threadIdx.x;
    int lane = tid & 31;
    int wave = tid >> 5;
    int hsel = lane >> 4;        // 0/1: which K pair this lane carries
    int l15  = lane & 15;

    int m0blk = blockIdx.y * 64;
    int n0blk = blockIdx.x * 64;

    // staging map: thread -> (row r0 + 16*p, 4 cols at c4) of the 64xKC tile
    int r0 = tid >> 3;           // 0..15
    int c4 = (tid & 7) * 4;      // 0,4,..,28

    const float* Ab = A  + (size_t)(m0blk + r0) * K + c4;
    const float* Bb = Wn + (size_t)(n0blk + r0) * K + c4;

    v4f aReg[4], bReg[4];
    v8f zero = {0.f,0.f,0.f,0.f,0.f,0.f,0.f,0.f};
    v8f acc[4];
#pragma unroll
    for (int t = 0; t < 4; ++t) acc[t] = zero;

    // ---- prologue: stage chunk 0 ----
#pragma unroll
    for (int p = 0; p < 4; ++p) {
        aReg[p] = *(const v4f*)(Ab + (size_t)(16 * p) * K);
        bReg[p] = *(const v4f*)(Bb + (size_t)(16 * p) * K);
    }
#pragma unroll
    for (int p = 0; p < 4; ++p) {
        *(v4f*)&ldsA[0][(r0 + 16 * p) * SRD + c4] = aReg[p];
        *(v4f*)&ldsB[0][(r0 + 16 * p) * SRD + c4] = bReg[p];
    }
    __syncthreads();

    int nchunks = K / KC;
    for (int ch = 0; ch < nchunks; ++ch) {
        int cur = ch & 1;
        int nxt = cur ^ 1;
        bool have_next = (ch + 1 < nchunks);

        // issue next chunk's global loads early (overlap with WMMAs below)
        if (have_next) {
            size_t ko = (size_t)(ch + 1) * KC;
#pragma unroll
            for (int p = 0; p < 4; ++p) {
                aReg[p] = *(const v4f*)(Ab + (size_t)(16 * p) * K + ko);
                bReg[p] = *(const v4f*)(Bb + (size_t)(16 * p) * K + ko);
            }
        }
        if (ch + 2 < nchunks) {   // gfx1250 global_prefetch_b8 for chunk+2
            size_t ko2 = (size_t)(ch + 2) * KC;
            __builtin_prefetch(Ab + ko2, 0, 1);
            __builtin_prefetch(Bb + ko2, 0, 1);
        }

        // ---- compute 32 WMMAs from the current LDS chunk ----
        const float* ap  = &ldsA[cur][(wave * 16 + l15) * SRD + 2 * hsel];
        const float* bp  = &ldsB[cur][l15 * SRD + 2 * hsel];
#pragma unroll
        for (int kk = 0; kk < KC; kk += 4) {
            v2f a  = *(const v2f*)(ap + kk);
            v2f b0 = *(const v2f*)(bp + kk);
            v2f b1 = *(const v2f*)(bp + 16 * SRD + kk);
            v2f b2 = *(const v2f*)(bp + 32 * SRD + kk);
            v2f b3 = *(const v2f*)(bp + 48 * SRD + kk);
            acc[0] = __builtin_amdgcn_wmma_f32_16x16x4_f32(false, a, false, b0, (short)0, acc[0], false, false);
            acc[1] = __builtin_amdgcn_wmma_f32_16x16x4_f32(false, a, false, b1, (short)0, acc[1], false, false);
            acc[2] = __builtin_amdgcn_wmma_f32_16x16x4_f32(false, a, false, b2, (short)0, acc[2], false, false);
            acc[3] = __builtin_amdgcn_wmma_f32_16x16x4_f32(false, a, false, b3, (short)0, acc[3], false, false);
        }

        if (have_next) {
#pragma unroll
            for (int p = 0; p < 4; ++p) {
                *(v4f*)&ldsA[nxt][(r0 + 16 * p) * SRD + c4] = aReg[p];
                *(v4f*)&ldsB[nxt][(r0 + 16 * p) * SRD + c4] = bReg[p];
            }
            __syncthreads();
        }
    }

    // ---- epilogue ----
    int m0 = m0blk + wave * 16;
#pragma unroll
    for (int t = 0; t < 4; ++t) {
#pragma unroll
        for (int v = 0; v < 8; ++v) {
            int row = m0 + v + (hsel << 3);
            int col = n0blk + t * 16 + l15;
            float pre = acc[t][v] + bias[col];
            size_t idx = (size_t)row * N + col;
            if (mode == 2) {
                out_act[idx] = pre;
            } else {
                out_act[idx] = tanhf(pre);
                float tg = tanh_lgrad(pre);
                out_grad[idx] = (mode == 0) ? (g0[col] + tg) : tg;
            }
        }
    }
}

// ---------------- Jacobian logsumexp for the 2048x2048 layer ----------------
// grad1[b,n] = LSE_c( g1[i,a,c] + grad0[b, i*32+c] ) + tg1[b,n],  n = i*32+a
__global__ void grad_lse1(const float* __restrict__ g1, const float* __restrict__ grad0,
                          const float* __restrict__ tg1, float* __restrict__ grad1) {
    int idx = blockIdx.x * blockDim.x + threadIdx.x;   // b*2048 + n
    int b = idx >> 11;
    int n = idx & 2047;
    int i = n >> 5;
    int a = n & 31;
    const float* gv = g1 + i * 1024 + a * 32;
    const float* pv = grad0 + (size_t)b * DH + (i << 5);
    float m = -3.0e38f;
#pragma unroll
    for (int c = 0; c < 32; ++c) m = fmaxf(m, gv[c] + pv[c]);
    float ssum = 0.0f;
#pragma unroll
    for (int c = 0; c < 32; ++c) ssum += expf(gv[c] + pv[c] - m);
    grad1[idx] = m + logf(ssum) + tg1[idx];
}

// ---------------- Per-flow epilogue: final LSE, gate, flip, log-det ----------------
// mode 0: first flow  (ldj  = sum)      mode 1: middle flow (ldj += sum)
// mode 2: last flow   (out[b] = ldj + sum(g - 0.5 y^2 - 0.5 log 2pi))
__global__ void flow_final(const float* __restrict__ g2, const float* __restrict__ grad1,
                           const float* __restrict__ out2, const float* __restrict__ x_in,
                           const float* __restrict__ gate, float* __restrict__ x_next,
                           float* __restrict__ ldj, float* __restrict__ final_out, int mode) {
    __shared__ float red[64];
    int b = blockIdx.x;
    int i = threadIdx.x;                 // 0..63
    const float* gv = g2 + i * 32;
    const float* pv = grad1 + (size_t)b * DH + (i << 5);
    float m = -3.0e38f;
#pragma unroll
    for (int c = 0; c < 32; ++c) m = fmaxf(m, gv[c] + pv[c]);
    float ssum = 0.0f;
#pragma unroll
    for (int c = 0; c < 32; ++c) ssum += expf(gv[c] + pv[c] - m);
    float gf = m + logf(ssum);

    float pre = out2[b * 64 + i];
    float contrib;
    if (mode < 2) {
        float gt = gate[0];
        float s = 1.0f / (1.0f + expf(-gt));
        float y = s * pre + (1.0f - s) * x_in[b * 64 + i];
        x_next[b * 64 + (63 - i)] = y;   // feature flip for next flow
        contrib = softplusf(gf + gt) - softplusf(gt);
    } else {
        contrib = gf - 0.5f * pre * pre - 0.5f * 1.8378770664093454836f; // log(2*pi)
    }
    red[i] = contrib;
    __syncthreads();
    for (int s2 = 32; s2 > 0; s2 >>= 1) { if (i < s2) red[i] += red[i + s2]; __syncthreads(); }
    if (i == 0) {
        float tot = red[0];
        if (mode == 0)      ldj[b] = tot;
        else if (mode == 1) ldj[b] += tot;
        else                final_out[b] = ldj[b] + tot;
    }
}

// ---------------- Host launcher ----------------
extern "C" void kernel_launch(void* const* d_in, const int* in_sizes, int n_in,
                              void* d_out, int out_size, void* d_ws, size_t ws_size,
                              hipStream_t stream) {
    (void)in_sizes; (void)n_in; (void)out_size; (void)ws_size;
    const float* x_input = (const float*)d_in[0];

    // Workspace layout (floats); total ~15.3M floats ~= 61 MB.
    float* ws = (float*)d_ws;
    float* wn0   = ws;              ws += 2048 * 64;
    float* wn1   = ws;              ws += 2048 * 2048;
    float* wn2   = ws;              ws += 64 * 2048;
    float* g0b   = ws;              ws += 2048;
    float* g1b   = ws;              ws += 64 * 32 * 32;
    float* g2b   = ws;              ws += 64 * 32;
    float* h0    = ws;              ws += (size_t)BV * DH;
    float* grad0 = ws;              ws += (size_t)BV * DH;
    float* h1    = ws;              ws += (size_t)BV * DH;
    float* tg1b  = ws;              ws += (size_t)BV * DH;
    float* grad1 = ws;              ws += (size_t)BV * DH;
    float* out2b = ws;              ws += (size_t)BV * DIMV;
    float* xa    = ws;              ws += (size_t)BV * DIMV;
    float* xb    = ws;              ws += (size_t)BV * DIMV;
    float* ldjb  = ws;              ws += BV;

    for (int f = 0; f < 3; ++f) {
        int base = 1 + f * 9;
        const float* W0 = (const float*)d_in[base + 0];
        const float* d0 = (const float*)d_in[base + 1];
        const float* b0 = (const float*)d_in[base + 2];
        const float* W1 = (const float*)d_in[base + 3];
        const float* d1 = (const float*)d_in[base + 4];
        const float* b1 = (const float*)d_in[base + 5];
        const float* W2 = (const float*)d_in[base + 6];
        const float* d2 = (const float*)d_in[base + 7];
        const float* b2 = (const float*)d_in[base + 8];

        prep_w0<<<2048, 64, 0, stream>>>(W0, d0, wn0, g0b);
        prep_w1<<<2048, 256, 0, stream>>>(W1, d1, wn1, g1b);
        prep_w2<<<64, 256, 0, stream>>>(W2, d2, wn2, g2b);

        const float* xin = (f == 0) ? x_input : ((f == 1) ? xa : xb);

        // layer 0: M=1024, N=2048, K=64  (tanh + grad init fused)
        wmma_gemm<<<dim3(2048 / 64, 1024 / 64), 128, 0, stream>>>(
            xin, wn0, b0, 64, 2048, 0, h0, grad0, g0b);
        // layer 1: M=1024, N=2048, K=2048 (tanh + tgrad fused)
        wmma_gemm<<<dim3(2048 / 64, 1024 / 64), 128, 0, stream>>>(
            h0, wn1, b1, 2048, 2048, 1, h1, tg1b, nullptr);
        // Jacobian chain for layer 1
        grad_lse1<<<(BV * DH) / 256, 256, 0, stream>>>(g1b, grad0, tg1b, grad1);
        // layer 2: M=1024, N=64, K=2048 (pre-activation only)
        wmma_gemm<<<dim3(64 / 64, 1024 / 64), 128, 0, stream>>>(
            h1, wn2, b2, 2048, 64, 2, out2b, nullptr, nullptr);

        const float* gate = (f == 0) ? (const float*)d_in[28]
                          : (f == 1) ? (const float*)d_in[29] : nullptr;
        float* xnext = (f == 0) ? xa : ((f == 1) ? xb : nullptr);
        flow_final<<<BV, 64, 0, stream>>>(g2b, grad1, out2b, xin, gate, xnext,
                                          ldjb, (f == 2) ? (float*)d_out : nullptr, f);
    }
}